// Cache_52192442581863
// MI455X (gfx1250) — compile-verified
//
#include <hip/hip_runtime.h>

// ---------------------------------------------------------------------------
// Problem constants (reference: B=4, S=2048, D=512, V=32000)
// ---------------------------------------------------------------------------
#define BB 4
#define SS 2048
#define DD 512
#define VV 32000
#define NT (BB * SS)          // 8192 tokens
#define BETA 0.9f
#define EPSF 1e-12f

typedef __attribute__((ext_vector_type(2))) float v2f;
typedef __attribute__((ext_vector_type(4))) float v4f;
typedef __attribute__((ext_vector_type(8))) float v8f;

// ---------------------------------------------------------------------------
// Kernel 1: zero workspace (segment sums [V*D] + counts [V]) and loss scalar.
// ---------------------------------------------------------------------------
__global__ __launch_bounds__(256)
void k_init(float* __restrict__ ws, float* __restrict__ loss_out, long n4) {
    long i = (long)blockIdx.x * blockDim.x + threadIdx.x;
    long stride = (long)gridDim.x * blockDim.x;
    v4f z = {};
    for (; i < n4; i += stride) ((v4f*)ws)[i] = z;
    if (blockIdx.x == 0 && threadIdx.x == 0) *loss_out = 0.0f;
}

// ---------------------------------------------------------------------------
// Kernel 2: fused loss (WMMA f32 16x16x4 Gram tile, diagonal = per-token dot)
//           + atomic segment-sum scatter of x * p[t, gold[t]].
// One wave32 owns 16 tokens; 256-thread block = 8 waves = 128 tokens;
// grid = NT/128 = 64 blocks (exact -> EXEC all-1s for every WMMA).
// ---------------------------------------------------------------------------
__global__ __launch_bounds__(256)
void k_fused(const float* __restrict__ x,
             const float* __restrict__ p,
             const int*   __restrict__ gold,
             const unsigned char* __restrict__ pad,
             const float* __restrict__ cache,
             float* __restrict__ loss_out,
             float* __restrict__ sums,
             float* __restrict__ counts) {
    const int lane  = threadIdx.x & 31;
    const int wave  = threadIdx.x >> 5;
    const int group = blockIdx.x * (blockDim.x >> 5) + wave;  // 16 tokens each
    const int t0    = group * 16;

    // -------- fragment indexing per ISA 32-bit A/B layouts (wave32) --------
    const int m    = lane & 15;          // token slot (M for A, N for B)
    const int kb   = (lane >> 4) * 2;    // K sub-offset: lanes 0-15 -> {0,1}, 16-31 -> {2,3}
    const int tokA = t0 + m;
    const int gA   = gold[tokA];

    const float* __restrict__ xrow = x     + (long)tokA * DD;
    const float* __restrict__ crow = cache + (long)gA   * DD;

    v8f acc = {};                 // 16x16 f32 accumulator (8 VGPRs)
    float ax2 = 0.0f;             // partial ||x_m||^2 (this lane's K slices)
    float bc2 = 0.0f;             // partial ||c_m||^2

    #pragma unroll 4
    for (int k = 0; k < DD; k += 4) {
        v2f a, b;
        a.x = xrow[k + kb];  a.y = xrow[k + kb + 1];
        b.x = crow[k + kb];  b.y = crow[k + kb + 1];
        ax2 += a.x * a.x + a.y * a.y;
        bc2 += b.x * b.x + b.y * b.y;
        // D = A(16x4) * B(4x16) + C : emits v_wmma_f32_16x16x4_f32
        acc = __builtin_amdgcn_wmma_f32_16x16x4_f32(
            /*neg_a=*/false, a, /*neg_b=*/false, b,
            /*c_mod=*/(short)0, acc, /*reuse_a=*/false, /*reuse_b=*/false);
    }

    // Combine the two half-wave K partitions: lane l and lane l^16 cover the
    // same token, complementary K slices.
    ax2 += __shfl_xor(ax2, 16, 32);
    bc2 += __shfl_xor(bc2, 16, 32);

    // Diagonal extraction: token n<8 -> lane n, VGPR n; token n>=8 -> lane
    // n+16, VGPR n-8. Lane holding token n also holds that token's ax2/bc2.
    float diag = 0.0f;
    #pragma unroll
    for (int r = 0; r < 8; ++r) {
        float e = acc[r];
        diag = (lane == r || lane == r + 24) ? e : diag;
    }
    const bool has = (lane < 8) || (lane >= 24);

    float lt = 0.0f;
    if (has) {
        float nx  = fmaxf(__builtin_sqrtf(ax2), EPSF);
        float nc  = fmaxf(__builtin_sqrtf(bc2), EPSF);
        float xn2 = ax2 / (nx * nx);
        float cn2 = bc2 / (nc * nc);
        float dn  = diag / (nx * nc);
        float keep = pad[tokA] ? 0.0f : 1.0f;
        lt = keep * (xn2 + cn2 - 2.0f * dn);
    }
    // wave32 reduction of the 16 diagonal contributions
    #pragma unroll
    for (int off = 16; off > 0; off >>= 1) lt += __shfl_xor(lt, off, 32);
    if (lane == 0) atomicAdd(loss_out, lt);

    // -------- scatter: contrib = keep * x[t] * p[t, gold[t]] --------------
    #pragma unroll 1
    for (int tk = 0; tk < 16; ++tk) {
        int   t    = t0 + tk;
        int   g    = gold[t];                      // uniform within wave loop
        float keep = pad[t] ? 0.0f : 1.0f;
        float pg   = p[(long)t * VV + g] * keep;   // single gathered scalar of p
        const float* __restrict__ xr = x + (long)t * DD;
        float* __restrict__ sr = sums + (long)g * DD;
        #pragma unroll
        for (int d = lane; d < DD; d += 32)
            atomicAdd(&sr[d], xr[d] * pg);
        if (lane == 0) atomicAdd(&counts[g], 1.0f);  // counts are NOT masked
    }
}

// ---------------------------------------------------------------------------
// Kernel 3: new_cache = count>0 ? beta*cache + (1-beta)*sums/max(count,1)
//                               : cache.  Vector loads, scalar stores
// (output starts at d_out+1 -> only 4B aligned).
// ---------------------------------------------------------------------------
__global__ __launch_bounds__(256)
void k_finalize(const float* __restrict__ cache,
                const float* __restrict__ sums,
                const float* __restrict__ counts,
                float* __restrict__ out) {
    long i  = (long)blockIdx.x * blockDim.x + threadIdx.x;   // one float4
    long n4 = (long)VV * DD / 4;
    if (i >= n4) return;
    long flat = i * 4;
    int  row  = (int)(flat >> 9);                            // /512
    float cnt = counts[row];
    v4f s = ((const v4f*)sums)[i];
    v4f c = ((const v4f*)cache)[i];
    float inv = 1.0f / fmaxf(cnt, 1.0f);
    v4f r = (cnt > 0.0f) ? (BETA * c + (1.0f - BETA) * (s * inv)) : c;
    out[flat + 0] = r.x;
    out[flat + 1] = r.y;
    out[flat + 2] = r.z;
    out[flat + 3] = r.w;
}

// ---------------------------------------------------------------------------
extern "C" void kernel_launch(void* const* d_in, const int* in_sizes, int n_in,
                              void* d_out, int out_size, void* d_ws, size_t ws_size,
                              hipStream_t stream) {
    const float* x     = (const float*)d_in[0];            // [8192, 512]
    const float* p     = (const float*)d_in[1];            // [8192, 32000]
    const int*   gold  = (const int*)d_in[2];              // [8192]
    const unsigned char* pad = (const unsigned char*)d_in[3]; // [8192] bool
    const float* cache = (const float*)d_in[4];            // [32000, 512]

    float* out  = (float*)d_out;       // out[0] = loss, out[1..] = new_cache_p
    float* sums = (float*)d_ws;        // [V*D] floats
    float* cnts = sums + (long)VV * DD; // [V] floats  (needs ~65.7 MB of ws)

    (void)in_sizes; (void)n_in; (void)out_size; (void)ws_size;

    // 1) zero sums+counts and the loss scalar
    long n4 = ((long)VV * DD + VV) / 4;
    k_init<<<2048, 256, 0, stream>>>(sums, out, n4);

    // 2) fused WMMA loss + atomic scatter: 512 waves, 8 waves/block
    k_fused<<<NT / 128, 256, 0, stream>>>(x, p, gold, pad, cache,
                                          out, sums, cnts);

    // 3) blend into d_out[1..]
    long fin4 = (long)VV * DD / 4;
    int  fblk = (int)((fin4 + 255) / 256);
    k_finalize<<<fblk, 256, 0, stream>>>(cache, sums, cnts, out + 1);
}